// OneGCN_74448963109327
// MI455X (gfx1250) — compile-verified
//
#include <hip/hip_runtime.h>

typedef float v2f __attribute__((ext_vector_type(2)));
typedef float v8f __attribute__((ext_vector_type(8)));

// ---------------------------------------------------------------------------
// Degree / normalization kernels
// ---------------------------------------------------------------------------
__global__ void deg_init_kernel(float* __restrict__ deg, int n) {
    int i = blockIdx.x * blockDim.x + threadIdx.x;
    if (i < n) deg[i] = 1.0f;  // self loop
}

__global__ void deg_count_kernel(const long long* __restrict__ dst,
                                 float* __restrict__ deg, long long E) {
    long long e = (long long)blockIdx.x * blockDim.x + threadIdx.x;
    if (e < E) {
        int d = (int)dst[e];
        unsafeAtomicAdd(&deg[d], 1.0f);  // hardware global_atomic_add_f32
    }
}

__global__ void deg_rsqrt_kernel(float* __restrict__ deg, int n) {
    int i = blockIdx.x * blockDim.x + threadIdx.x;
    if (i < n) deg[i] = rsqrtf(deg[i]);
}

// ---------------------------------------------------------------------------
// fp32 WMMA GEMM: C[M,N] = A[M,K] @ B[K,N] (+ bias)
// One wave computes a 32x64 C tile: 2 M-subtiles x 4 N-subtiles of
// v_wmma_f32_16x16x4_f32, sharing the B fragments across both M-subtiles
// (8 WMMA per 10 loads -> ~10.7 FLOP/byte from L2).
// Requires: M%32==0 (whole-wave early-out keeps EXEC all-ones), N%64==0, K%4==0.
// ---------------------------------------------------------------------------
__global__ __launch_bounds__(256)
void gemm_wmma_f32_kernel(const float* __restrict__ A, const float* __restrict__ B,
                          const float* __restrict__ bias, float* __restrict__ C,
                          int M, int K, int N, int add_bias) {
    const int lane = threadIdx.x & 31;
    const int wave = threadIdx.x >> 5;
    const int half = lane >> 4;   // 0: lanes 0-15, 1: lanes 16-31
    const int l15  = lane & 15;

    const int mb = (blockIdx.x * 8 + wave) * 32;   // 32-row tile base
    if (mb >= M) return;                            // whole wave exits uniformly
    const int nb = blockIdx.y * 64;                 // 64-col tile base

    v8f accA0 = {}; v8f accA1 = {}; v8f accA2 = {}; v8f accA3 = {};  // rows mb..mb+15
    v8f accB0 = {}; v8f accB1 = {}; v8f accB2 = {}; v8f accB3 = {};  // rows mb+16..mb+31

    const float* __restrict__ Arow0 = A + (size_t)(mb + l15) * K;
    const float* __restrict__ Arow1 = Arow0 + (size_t)16 * K;

    for (int k0 = 0; k0 < K; k0 += 4) {
        const int ks = k0 + 2 * half;
        // A 16x4: lanes 0-15 hold K = k0+{0,1}; lanes 16-31 hold K = k0+{2,3}
        v2f a0 = *(const v2f*)(Arow0 + ks);
        v2f a1 = *(const v2f*)(Arow1 + ks);
        // B 4x16: VGPR0 row = ks, VGPR1 row = ks+1; N striped over lanes 0-15
        const float* __restrict__ Bp = B + (size_t)ks * N + (nb + l15);
        v2f b0, b1, b2, b3;
        b0[0] = Bp[0];  b0[1] = Bp[N];
        b1[0] = Bp[16]; b1[1] = Bp[N + 16];
        b2[0] = Bp[32]; b2[1] = Bp[N + 32];
        b3[0] = Bp[48]; b3[1] = Bp[N + 48];
        accA0 = __builtin_amdgcn_wmma_f32_16x16x4_f32(false, a0, false, b0, (short)0, accA0, false, false);
        accB0 = __builtin_amdgcn_wmma_f32_16x16x4_f32(false, a1, false, b0, (short)0, accB0, false, false);
        accA1 = __builtin_amdgcn_wmma_f32_16x16x4_f32(false, a0, false, b1, (short)0, accA1, false, false);
        accB1 = __builtin_amdgcn_wmma_f32_16x16x4_f32(false, a1, false, b1, (short)0, accB1, false, false);
        accA2 = __builtin_amdgcn_wmma_f32_16x16x4_f32(false, a0, false, b2, (short)0, accA2, false, false);
        accB2 = __builtin_amdgcn_wmma_f32_16x16x4_f32(false, a1, false, b2, (short)0, accB2, false, false);
        accA3 = __builtin_amdgcn_wmma_f32_16x16x4_f32(false, a0, false, b3, (short)0, accA3, false, false);
        accB3 = __builtin_amdgcn_wmma_f32_16x16x4_f32(false, a1, false, b3, (short)0, accB3, false, false);
    }

    // C layout: VGPR v -> row base + v + 8*half, col nb + 16*sub + l15
    float bv0 = 0.f, bv1 = 0.f, bv2 = 0.f, bv3 = 0.f;
    if (add_bias) {
        bv0 = bias[nb + l15];
        bv1 = bias[nb + 16 + l15];
        bv2 = bias[nb + 32 + l15];
        bv3 = bias[nb + 48 + l15];
    }
    const int row0 = mb + 8 * half;
#pragma unroll
    for (int v = 0; v < 8; ++v) {
        float* crow = C + (size_t)(row0 + v) * N + nb + l15;
        crow[0]  = accA0[v] + bv0;
        crow[16] = accA1[v] + bv1;
        crow[32] = accA2[v] + bv2;
        crow[48] = accA3[v] + bv3;
        float* crow2 = crow + (size_t)16 * N;
        crow2[0]  = accB0[v] + bv0;
        crow2[16] = accB1[v] + bv1;
        crow2[32] = accB2[v] + bv2;
        crow2[48] = accB3[v] + bv3;
    }
}

// ---------------------------------------------------------------------------
// Aggregation: out[i] = XW[i] * inv_sqrt[i]^2 (self loop term, also initializes out)
// ---------------------------------------------------------------------------
__global__ void agg_init_kernel(const float* __restrict__ XW,
                                const float* __restrict__ invs,
                                float* __restrict__ out, int n, int F) {
    long long i = (long long)blockIdx.x * blockDim.x + threadIdx.x;
    const int fc4 = F >> 2;
    long long total = (long long)n * fc4;
    if (i >= total) return;
    int node = (int)(i / fc4);
    float w = invs[node] * invs[node];
    float4 v = ((const float4*)XW)[i];
    v.x *= w; v.y *= w; v.z *= w; v.w *= w;
    ((float4*)out)[i] = v;
}

// ---------------------------------------------------------------------------
// Aggregation: per-edge gather + scale + scatter-add (float4 per thread)
// ---------------------------------------------------------------------------
__global__ void agg_edges_kernel(const float* __restrict__ XW,
                                 const long long* __restrict__ src,
                                 const long long* __restrict__ dst,
                                 const float* __restrict__ invs,
                                 float* __restrict__ out, long long E, int F) {
    long long t = (long long)blockIdx.x * blockDim.x + threadIdx.x;
    const int fc4 = F >> 2;
    long long total = E * fc4;
    if (t >= total) return;
    long long e = t / fc4;
    int c = (int)(t - e * fc4);
    int s = (int)src[e];
    int d = (int)dst[e];
    float w = invs[s] * invs[d];
    float4 v = ((const float4*)(XW + (size_t)s * F))[c];
    float* o = out + (size_t)d * F + (size_t)c * 4;
    unsafeAtomicAdd(o + 0, v.x * w);
    unsafeAtomicAdd(o + 1, v.y * w);
    unsafeAtomicAdd(o + 2, v.z * w);
    unsafeAtomicAdd(o + 3, v.w * w);
}

// ---------------------------------------------------------------------------
// Bias + ReLU (in place), float4
// ---------------------------------------------------------------------------
__global__ void bias_relu_kernel(float* __restrict__ H, const float* __restrict__ b,
                                 int n, int F) {
    long long i = (long long)blockIdx.x * blockDim.x + threadIdx.x;
    const int fc4 = F >> 2;
    long long total = (long long)n * fc4;
    if (i >= total) return;
    int c4 = (int)(i % fc4) * 4;
    float4 h = ((float4*)H)[i];
    h.x = fmaxf(h.x + b[c4 + 0], 0.f);
    h.y = fmaxf(h.y + b[c4 + 1], 0.f);
    h.z = fmaxf(h.z + b[c4 + 2], 0.f);
    h.w = fmaxf(h.w + b[c4 + 3], 0.f);
    ((float4*)H)[i] = h;
}

// ---------------------------------------------------------------------------
// Host-side orchestration
// ---------------------------------------------------------------------------
extern "C" void kernel_launch(void* const* d_in, const int* in_sizes, int n_in,
                              void* d_out, int out_size, void* d_ws, size_t ws_size,
                              hipStream_t stream) {
    const float*     x   = (const float*)d_in[0];
    const long long* ei  = (const long long*)d_in[1];
    const float*     W1  = (const float*)d_in[2];
    const float*     b1  = (const float*)d_in[3];
    const float*     W2  = (const float*)d_in[4];
    const float*     b2  = (const float*)d_in[5];
    const float*     W3  = (const float*)d_in[6];
    const float*     b3  = (const float*)d_in[7];
    const float*     Wc  = (const float*)d_in[8];
    const float*     bc  = (const float*)d_in[9];

    const int n = in_sizes[0] / 256;          // 20000 nodes
    const long long E = in_sizes[1] / 2;      // 320000 edges
    const long long* src = ei;
    const long long* dst = ei + E;

    // Workspace layout: inv_sqrt[n] | bufA[n*1024] | bufB[n*1024]
    float* invs = (float*)d_ws;
    size_t invs_pad = ((size_t)n * sizeof(float) + 255) & ~(size_t)255;
    float* bufA = (float*)((char*)d_ws + invs_pad);
    float* bufB = bufA + (size_t)n * 1024;

    const int TB = 256;

    // --- degree normalization ---
    deg_init_kernel<<<(n + TB - 1) / TB, TB, 0, stream>>>(invs, n);
    deg_count_kernel<<<(unsigned)((E + TB - 1) / TB), TB, 0, stream>>>(dst, invs, E);
    deg_rsqrt_kernel<<<(n + TB - 1) / TB, TB, 0, stream>>>(invs, n);

    auto launch_gemm = [&](const float* A, const float* W, const float* bias,
                           float* C, int K, int F, int add_bias) {
        dim3 grid((n + 255) / 256, F / 64);   // 8 waves x 32 rows = 256 rows / block
        gemm_wmma_f32_kernel<<<grid, 256, 0, stream>>>(A, W, bias, C, n, K, F, add_bias);
    };
    auto launch_agg = [&](const float* XW, float* out, const float* bias, int F) {
        long long tot_n = (long long)n * (F / 4);
        long long tot_e = E * (F / 4);
        agg_init_kernel<<<(unsigned)((tot_n + TB - 1) / TB), TB, 0, stream>>>(XW, invs, out, n, F);
        agg_edges_kernel<<<(unsigned)((tot_e + TB - 1) / TB), TB, 0, stream>>>(XW, src, dst, invs, out, E, F);
        bias_relu_kernel<<<(unsigned)((tot_n + TB - 1) / TB), TB, 0, stream>>>(out, bias, n, F);
    };

    // Layer 1: x[n,256] -> bufA = x@W1 [n,1024] -> bufB aggregated+bias+relu
    launch_gemm(x, W1, b1, bufA, 256, 1024, 0);
    launch_agg(bufA, bufB, b1, 1024);

    // Layer 2: bufB[n,1024] -> bufA = H1@W2 [n,512] -> bufB aggregated
    launch_gemm(bufB, W2, b2, bufA, 1024, 512, 0);
    launch_agg(bufA, bufB, b2, 512);

    // Layer 3: bufB[n,512] -> bufA = H2@W3 [n,128] -> bufB aggregated
    launch_gemm(bufB, W3, b3, bufA, 512, 128, 0);
    launch_agg(bufA, bufB, b3, 128);

    // Classifier: bufB[n,128] @ Wc [128,64] + bc -> d_out (no relu, no aggregation)
    launch_gemm(bufB, Wc, bc, (float*)d_out, 128, 64, 1);
}